// Discriminator_15668040696127
// MI455X (gfx1250) — compile-verified
//
#include <hip/hip_runtime.h>
#include <math.h>

// Problem constants from the reference.
#define BB   8192
#define DD   4096
#define NBLK 1024          // 8 waves/block * 1024 blocks = 8192 waves = 1 row/wave
#define TPB  256
#define WPB  (TPB / 32)
#define EPSV 1e-6f

typedef __attribute__((ext_vector_type(2))) float v2f;
typedef __attribute__((ext_vector_type(4))) float v4f;
typedef __attribute__((ext_vector_type(8))) float v8f;

#if defined(__has_builtin)
#  if __has_builtin(__builtin_amdgcn_wmma_f32_16x16x4_f32)
#    define HAVE_WMMA_F32X4 1
#  endif
#endif

__global__ __launch_bounds__(TPB)
void disc_rows_kernel(const float* __restrict__ samples,
                      const float* __restrict__ Dv1,
                      const float* __restrict__ Dv2,
                      float* __restrict__ block_part) {
    __shared__ float swacc[WPB];

    const int lane = threadIdx.x & 31;
    const int wave = threadIdx.x >> 5;
    const int wave_global = blockIdx.x * WPB + wave;
    const int total_waves = gridDim.x * WPB;

    float wacc = 0.0f;

    for (int row = wave_global; row < BB; row += total_waves) {
        const v4f* __restrict__ qp =
            (const v4f*)(samples + (size_t)(row * 3 + 0) * DD);
        const v4f* __restrict__ ap =
            (const v4f*)(samples + (size_t)(row * 3 + 1) * DD);
        const v4f* __restrict__ dp =
            (const v4f*)(samples + (size_t)(row * 3 + 2) * DD);

        float qd = 0.f, ad = 0.f, qq = 0.f, aa = 0.f, dd = 0.f;

        // D/ (32 lanes * 4 floats) = 32 iterations; read-once -> non-temporal.
        #pragma unroll 4
        for (int it = 0; it < (DD / 128); ++it) {
            const int idx = lane + (it << 5);
            if (it + 8 < (DD / 128)) {
                __builtin_prefetch((const void*)(qp + idx + 256), 0, 0);
                __builtin_prefetch((const void*)(ap + idx + 256), 0, 0);
                __builtin_prefetch((const void*)(dp + idx + 256), 0, 0);
            }
            v4f qv = __builtin_nontemporal_load(qp + idx);
            v4f av = __builtin_nontemporal_load(ap + idx);
            v4f dv = __builtin_nontemporal_load(dp + idx);
            #pragma unroll
            for (int c = 0; c < 4; ++c) {
                qd = fmaf(qv[c], dv[c], qd);
                ad = fmaf(av[c], dv[c], ad);
                qq = fmaf(qv[c], qv[c], qq);
                aa = fmaf(av[c], av[c], aa);
                dd = fmaf(dv[c], dv[c], dd);
            }
        }

        // ---- cross-lane reduction of 5 accumulators ----
        // Stage 1: reduce within groups of 8 lanes (xor 1,2,4).
        #pragma unroll
        for (int off = 1; off <= 4; off <<= 1) {
            qd += __shfl_xor(qd, off, 32);
            ad += __shfl_xor(ad, off, 32);
            qq += __shfl_xor(qq, off, 32);
            aa += __shfl_xor(aa, off, 32);
            dd += __shfl_xor(dd, off, 32);
        }

#if defined(HAVE_WMMA_F32X4)
        // Stage 2: one V_WMMA_F32_16X16X4_F32 with B = ones sums the four
        // group-of-8 partials per quantity: D[m][n] = sum_k A[m][k].
        // A (16x4 f32) layout: lanes 0-15 hold K0 (v0)/K1 (v1) of row M=lane,
        // lanes 16-31 hold K2/K3.  Map K -> lane-group: K0<-lanes0-7,
        // K1<-lanes8-15, K2<-lanes16-23, K3<-lanes24-31.
        const int src0 = lane & 16;        // group sums for K0 / K2
        const int src1 = (lane & 16) | 8;  // group sums for K1 / K3
        const float b00 = __shfl(qd, src0, 32), b10 = __shfl(qd, src1, 32);
        const float b01 = __shfl(ad, src0, 32), b11 = __shfl(ad, src1, 32);
        const float b02 = __shfl(qq, src0, 32), b12 = __shfl(qq, src1, 32);
        const float b03 = __shfl(aa, src0, 32), b13 = __shfl(aa, src1, 32);
        const float b04 = __shfl(dd, src0, 32), b14 = __shfl(dd, src1, 32);

        const int rowm = lane & 15;        // A-matrix row index M
        float A0 = (rowm == 0) ? b00 : (rowm == 1) ? b01 : (rowm == 2) ? b02
                 : (rowm == 3) ? b03 : (rowm == 4) ? b04 : 0.0f;
        float A1 = (rowm == 0) ? b10 : (rowm == 1) ? b11 : (rowm == 2) ? b12
                 : (rowm == 3) ? b13 : (rowm == 4) ? b14 : 0.0f;

        v2f amat; amat[0] = A0;   amat[1] = A1;
        v2f bmat; bmat[0] = 1.0f; bmat[1] = 1.0f;
        v8f cm = {};
        cm = __builtin_amdgcn_wmma_f32_16x16x4_f32(
                 false, amat, false, bmat, (short)0, cm, false, false);

        // D row m sits in VGPR m for lanes 0-15; broadcast from lane 0.
        qd = __shfl(cm[0], 0, 32);
        ad = __shfl(cm[1], 0, 32);
        qq = __shfl(cm[2], 0, 32);
        aa = __shfl(cm[3], 0, 32);
        dd = __shfl(cm[4], 0, 32);
#else
        // Fallback: finish the butterfly (xor 8, 16).
        #pragma unroll
        for (int off = 8; off <= 16; off <<= 1) {
            qd += __shfl_xor(qd, off, 32);
            ad += __shfl_xor(ad, off, 32);
            qq += __shfl_xor(qq, off, 32);
            aa += __shfl_xor(aa, off, 32);
            dd += __shfl_xor(dd, off, 32);
        }
#endif

        const float cos1 = qd / fmaxf(sqrtf(qq) * sqrtf(dd), EPSV);
        const float cos2 = ad / fmaxf(sqrtf(aa) * sqrtf(dd), EPSV);
        wacc += cos1 * Dv1[row] + cos2 * Dv2[row];   // uniform across lanes
    }

    if (lane == 0) swacc[wave] = wacc;
    __syncthreads();
    if (threadIdx.x == 0) {
        float s = 0.0f;
        #pragma unroll
        for (int i = 0; i < WPB; ++i) s += swacc[i];
        block_part[blockIdx.x] = s;
    }
}

__global__ __launch_bounds__(256)
void disc_finalize_kernel(const float* __restrict__ block_part,
                          const float* __restrict__ labels,
                          float* __restrict__ out) {
    __shared__ float s[256];
    float v = 0.0f;
    #pragma unroll
    for (int i = threadIdx.x; i < NBLK; i += 256) v += block_part[i];
    s[threadIdx.x] = v;
    __syncthreads();
    for (int off = 128; off > 0; off >>= 1) {
        if (threadIdx.x < off) s[threadIdx.x] += s[threadIdx.x + off];
        __syncthreads();
    }
    if (threadIdx.x == 0) {
        const float score = s[0];
        const float y = labels[0];
        const float bce = fmaxf(score, 0.0f) - score * y
                        + log1pf(expf(-fabsf(score)));
        out[0] = bce;   // mean over a single element
    }
}

extern "C" void kernel_launch(void* const* d_in, const int* in_sizes, int n_in,
                              void* d_out, int out_size, void* d_ws, size_t ws_size,
                              hipStream_t stream) {
    (void)in_sizes; (void)n_in; (void)out_size; (void)ws_size;
    const float* samples = (const float*)d_in[0];
    const float* labels  = (const float*)d_in[1];
    const float* Dv1     = (const float*)d_in[2];
    const float* Dv2     = (const float*)d_in[3];
    float* part = (float*)d_ws;       // NBLK floats of scratch
    float* out  = (float*)d_out;

    disc_rows_kernel<<<NBLK, TPB, 0, stream>>>(samples, Dv1, Dv2, part);
    disc_finalize_kernel<<<1, 256, 0, stream>>>(part, labels, out);
}